// radialNO3d_41661182771331
// MI455X (gfx1250) — compile-verified
//
#include <hip/hip_runtime.h>
#include <hip/hip_bf16.h>
#include <math.h>

// ---------------------------------------------------------------------------
// Radial FNO-3D forward for MI455X (gfx1250, wave32, WMMA).
// Pointwise channel GEMMs use V_WMMA_F32_16X16X4_F32 (true f32 matrix core).
// Spectral conv implemented as truncated DFT (only the kept 23x23x8 modes are
// ever computed), with the radial weight construction done as an index decode.
// ---------------------------------------------------------------------------

typedef float v2f __attribute__((ext_vector_type(2)));
typedef float v8f __attribute__((ext_vector_type(8)));

#define NTOT   327680      // B*S1*S2*S3 = 2*64*64*40
#define S1D    64
#define S2D    64
#define S3D    40
#define BD     2
#define WIDTH_ 64
#define MODES_ 12
#define TMOD_  8
#define MX23   23          // 2*MODES-1
#define NMODE  4232        // 23*23*8
#define TWO_PI 6.28318530717958647692f

// ===========================================================================
// WMMA GEMM: Y[M x ncols] = act( W[MxK=64] * X + bias (+ R) )
// M = MT*16. One wave -> M x 16 tile. 8 waves / block -> 128 columns / block.
// f32 WMMA 16x16x4: A (16x4) = v2f, B (4x16) = v2f, C/D (16x16) = v8f.
//   A frag: lanes 0-15 hold K = kb,kb+1 ; lanes 16-31 hold K = kb+2,kb+3.
//   C frag: vgpr j: lanes 0-15 -> M = 16m+j ; lanes 16-31 -> M = 16m+8+j.
// act: 0 = none, 1 = relu, 2 = exact gelu
// All global streaming addresses are strength-reduced to pointer increments.
// ===========================================================================
template <int MT>
__global__ __launch_bounds__(256) void gemm_wmma_k64(
    const float* __restrict__ W, const float* __restrict__ bias,
    const float* __restrict__ X, long ldx,
    const float* __restrict__ R, long ldr,
    float* __restrict__ Y, long ldy,
    int ncols, int act)
{
    constexpr int K = 64;
    constexpr int M = MT * 16;
    __shared__ float sW[M * K];
    for (int i = threadIdx.x; i < M * K; i += 256) sW[i] = W[i];
    __syncthreads();

    const int wave = threadIdx.x >> 5;
    const int lane = threadIdx.x & 31;
    const int n0 = (blockIdx.x * 8 + wave) * 16;
    if (n0 >= ncols) return;
    const int lh  = lane >> 4;   // half-wave select
    const int l15 = lane & 15;

    v8f acc[MT];
#pragma unroll
    for (int m = 0; m < MT; ++m)
#pragma unroll
        for (int j = 0; j < 8; ++j) acc[m][j] = 0.0f;

    // Column base pointer for this lane; K-loop walks rows by pointer steps.
    const float* xp = X + (long)(lh * 2) * ldx + n0 + l15;
    const long   xstep = 4 * ldx;

#pragma unroll 4
    for (int kb = 0; kb < K; kb += 4) {
        v2f bf;
        bf.x = xp[0];
        bf.y = xp[ldx];
        if (kb + 4 < K)   // stream hint for the next K-slab of X
            __builtin_prefetch(xp + xstep, 0, 1);
        const int kbase = kb + lh * 2;
#pragma unroll
        for (int m = 0; m < MT; ++m) {
            v2f af;
            af.x = sW[(m * 16 + l15) * K + kbase + 0];
            af.y = sW[(m * 16 + l15) * K + kbase + 1];
            acc[m] = __builtin_amdgcn_wmma_f32_16x16x4_f32(
                false, af, false, bf, (short)0, acc[m], false, false);
        }
        xp += xstep;
    }

    // Epilogue: incremental row pointers (no per-element index multiplies).
#pragma unroll
    for (int m = 0; m < MT; ++m) {
        const int r0 = m * 16 + lh * 8;
        float* yp = Y + (long)r0 * ldy + n0 + l15;
        const float* rp = R ? (R + (long)r0 * ldr + n0 + l15) : nullptr;
#pragma unroll
        for (int j = 0; j < 8; ++j) {
            float v = acc[m][j] + bias[r0 + j];
            if (rp) { v += *rp; rp += ldr; }
            if (act == 1)      v = fmaxf(v, 0.0f);
            else if (act == 2) v = 0.5f * v * (1.0f + erff(v * 0.70710678118654752f));
            *yp = v;
            yp += ldy;
        }
    }
}

// ===========================================================================
// Lift: concat grid coords, project 13 -> 64 channels. out layout [c][n].
// ===========================================================================
__global__ void lift_kernel(const float* __restrict__ xin,
                            const float* __restrict__ pw,
                            const float* __restrict__ pb,
                            float* __restrict__ out)
{
    const int n = blockIdx.x * blockDim.x + threadIdx.x;
    if (n >= NTOT) return;
    const int t = n % S3D;
    int tmp = n / S3D;
    const int y = tmp % S2D; tmp /= S2D;
    const int x = tmp % S1D;
    float f[13];
#pragma unroll
    for (int i = 0; i < 10; ++i) f[i] = xin[(long)n * 10 + i];
    f[10] = (float)x * (1.0f / 63.0f);
    f[11] = (float)y * (1.0f / 63.0f);
    f[12] = (float)t * (1.0f / 39.0f);
    float* op = out + n;
    for (int c = 0; c < WIDTH_; ++c) {
        float acc = pb[c];
#pragma unroll
        for (int i = 0; i < 13; ++i) acc += pw[c * 13 + i] * f[i];
        *op = acc;
        op += NTOT;
    }
}

// ===========================================================================
// Forward truncated DFT along z (rfft, keep kz=0..7). Thread per (c,b,x,y).
// ===========================================================================
__global__ void dftz_fwd(const float* __restrict__ A,
                         float* __restrict__ ZR, float* __restrict__ ZI)
{
    const int id = blockIdx.x * blockDim.x + threadIdx.x;
    if (id >= WIDTH_ * BD * S1D * S2D) return;
    const int c   = id / (BD * S1D * S2D);
    const int rem = id % (BD * S1D * S2D);
    const long bi = (long)c * NTOT + (long)rem * S3D;
    const long bo = (long)c * (BD * S1D * S2D * TMOD_) + (long)rem * TMOD_;
    float re[TMOD_], im[TMOD_];
#pragma unroll
    for (int k = 0; k < TMOD_; ++k) { re[k] = 0.0f; im[k] = 0.0f; }
    for (int tt = 0; tt < S3D; ++tt) {
        const float v = A[bi + tt];
#pragma unroll
        for (int k = 0; k < TMOD_; ++k) {
            float s, co;
            __sincosf(-(TWO_PI / S3D) * (float)(k * tt), &s, &co);
            re[k] += v * co; im[k] += v * s;
        }
    }
#pragma unroll
    for (int k = 0; k < TMOD_; ++k) { ZR[bo + k] = re[k]; ZI[bo + k] = im[k]; }
}

// ===========================================================================
// Forward DFT along y: 64 -> 23 shifted modes (m = my-11). Thread per (c,b,x,kz).
// ===========================================================================
__global__ void dfty_fwd(const float* __restrict__ ZR, const float* __restrict__ ZI,
                         float* __restrict__ YR, float* __restrict__ YI)
{
    const int id = blockIdx.x * blockDim.x + threadIdx.x;
    if (id >= WIDTH_ * BD * S1D * TMOD_) return;
    const int c  = id / (BD * S1D * TMOD_);
    const int r  = id % (BD * S1D * TMOD_);
    const int bx = r / TMOD_;     // b*64 + x
    const int kz = r % TMOD_;
    const long bi = (long)c * (BD * S1D * S2D * TMOD_) + (long)bx * (S2D * TMOD_) + kz;
    const long bo = (long)c * (BD * S1D * MX23 * TMOD_) + (long)bx * (MX23 * TMOD_) + kz;
    float yr[MX23], yi[MX23];
#pragma unroll
    for (int m = 0; m < MX23; ++m) { yr[m] = 0.0f; yi[m] = 0.0f; }
    for (int y = 0; y < S2D; ++y) {
        const float zr = ZR[bi + (long)y * TMOD_];
        const float zi = ZI[bi + (long)y * TMOD_];
#pragma unroll
        for (int m = 0; m < MX23; ++m) {
            float s, co;
            __sincosf((TWO_PI / S2D) * (float)((m - 11) * y), &s, &co);
            yr[m] += zr * co + zi * s;   // e^{-i theta}
            yi[m] += zi * co - zr * s;
        }
    }
#pragma unroll
    for (int m = 0; m < MX23; ++m) {
        YR[bo + (long)m * TMOD_] = yr[m];
        YI[bo + (long)m * TMOD_] = yi[m];
    }
}

// ===========================================================================
// Forward DFT along x: 64 -> 23. Thread per (c,b,my,kz). Output xs[i][b][mx][my][kz].
// ===========================================================================
__global__ void dftx_fwd(const float* __restrict__ YR, const float* __restrict__ YI,
                         float* __restrict__ XSR, float* __restrict__ XSI)
{
    const int id = blockIdx.x * blockDim.x + threadIdx.x;
    if (id >= WIDTH_ * BD * MX23 * TMOD_) return;
    const int c  = id / (BD * MX23 * TMOD_);
    const int r  = id % (BD * MX23 * TMOD_);
    const int b  = r / (MX23 * TMOD_);
    const int r2 = r % (MX23 * TMOD_);
    const int my = r2 / TMOD_;
    const int kz = r2 % TMOD_;
    const long bi = (long)c * (BD * S1D * MX23 * TMOD_) + (long)b * (S1D * MX23 * TMOD_)
                  + (long)my * TMOD_ + kz;
    float xr[MX23], xi[MX23];
#pragma unroll
    for (int m = 0; m < MX23; ++m) { xr[m] = 0.0f; xi[m] = 0.0f; }
    for (int x = 0; x < S1D; ++x) {
        const float yr = YR[bi + (long)x * (MX23 * TMOD_)];
        const float yi = YI[bi + (long)x * (MX23 * TMOD_)];
#pragma unroll
        for (int m = 0; m < MX23; ++m) {
            float s, co;
            __sincosf((TWO_PI / S1D) * (float)((m - 11) * x), &s, &co);
            xr[m] += yr * co + yi * s;
            xi[m] += yi * co - yr * s;
        }
    }
    const long bo = (long)c * (BD * NMODE) + (long)b * NMODE + (long)my * TMOD_ + kz;
#pragma unroll
    for (int m = 0; m < MX23; ++m) {
        XSR[bo + (long)m * (MX23 * TMOD_)] = xr[m];
        XSI[bo + (long)m * (MX23 * TMOD_)] = xi[m];
    }
}

// ===========================================================================
// Radial weight decode: W[i,o,a,b,kz] -> W_LC[i,o,ia,0,kz] or W_LR[i,o,ia,ib,kz]
// (pure index remap of the concat/flip/rot90 construction; i,o pass through).
// ===========================================================================
__device__ __forceinline__ void decode_mode(int a, int b, int& src, int& ia, int& ib)
{
    int a2, b2;
    if (b < 11) { a2 = 22 - a; b2 = 11 - b; } else { a2 = a; b2 = b - 11; }
    int u, v;
    if (a2 < 12) {
        if (b2 == 0) { src = 0; ia = 11 - a2; ib = 0; return; }  // flip(W_LC)
        u = b2 - 1; v = 11 - a2;                                  // rot90(W_LRf)
    } else {
        u = a2 - 12; v = b2;                                      // W_LRf
    }
    if (v == 0) { src = 0; ia = u + 1; ib = 0; }
    else        { src = 1; ia = u;     ib = v - 1; }
}

// Mode einsum: S[b,o,m] = sum_i W[i,o,m] * xs[b,i,m]  (W real, xs complex)
__global__ void mode_einsum(const float* __restrict__ XSR, const float* __restrict__ XSI,
                            const float* __restrict__ WLC, const float* __restrict__ WLR,
                            float* __restrict__ SR, float* __restrict__ SI)
{
    const int id = blockIdx.x * blockDim.x + threadIdx.x;
    if (id >= BD * WIDTH_ * NMODE) return;
    const int m = id % NMODE;
    const int o = (id / NMODE) % WIDTH_;
    const int b = id / (NMODE * WIDTH_);
    const int kz = m % TMOD_;
    const int my = (m / TMOD_) % MX23;
    const int mx = m / (MX23 * TMOD_);
    int src, ia, ib;
    decode_mode(mx, my, src, ia, ib);
    const float* wp;
    long istride;
    if (src == 0) { wp = WLC + ((long)o * MODES_ + ia) * TMOD_ + kz;
                    istride = (long)WIDTH_ * MODES_ * TMOD_; }
    else          { wp = WLR + (((long)o * 11 + ia) * 11 + ib) * TMOD_ + kz;
                    istride = (long)WIDTH_ * 11 * 11 * TMOD_; }
    float sr = 0.0f, si = 0.0f;
    const float* xrp = XSR + (long)b * NMODE + m;
    const float* xip = XSI + (long)b * NMODE + m;
    for (int i = 0; i < WIDTH_; ++i) {
        const float w = *wp;
        sr += w * (*xrp);
        si += w * (*xip);
        wp  += istride;
        xrp += BD * NMODE;
        xip += BD * NMODE;
    }
    SR[(long)o * (BD * NMODE) + (long)b * NMODE + m] = sr;
    SI[(long)o * (BD * NMODE) + (long)b * NMODE + m] = si;
}

// ===========================================================================
// Inverse DFT x: 23 modes -> 64. Thread per (b,o,x,my,kz).
// ===========================================================================
__global__ void idftx(const float* __restrict__ SR, const float* __restrict__ SI,
                      float* __restrict__ T1R, float* __restrict__ T1I)
{
    const int id = blockIdx.x * blockDim.x + threadIdx.x;
    if (id >= BD * WIDTH_ * S1D * MX23 * TMOD_) return;
    const int kz = id % TMOD_;
    const int my = (id / TMOD_) % MX23;
    const int x  = (id / (MX23 * TMOD_)) % S1D;
    const int o  = (id / (S1D * MX23 * TMOD_)) % WIDTH_;
    const int b  = id / (WIDTH_ * S1D * MX23 * TMOD_);
    const float* srp = SR + (long)o * (BD * NMODE) + (long)b * NMODE + (long)my * TMOD_ + kz;
    const float* sip = SI + (long)o * (BD * NMODE) + (long)b * NMODE + (long)my * TMOD_ + kz;
    float tr = 0.0f, ti = 0.0f;
#pragma unroll
    for (int mx = 0; mx < MX23; ++mx) {
        float s, co;
        __sincosf((TWO_PI / S1D) * (float)((mx - 11) * x), &s, &co);
        const float sr = srp[(long)mx * (MX23 * TMOD_)];
        const float si = sip[(long)mx * (MX23 * TMOD_)];
        tr += sr * co - si * s;        // e^{+i theta}
        ti += si * co + sr * s;
    }
    const long bo = (long)o * (BD * S1D * MX23 * TMOD_)
                  + ((long)(b * S1D + x) * MX23 + my) * TMOD_ + kz;
    T1R[bo] = tr; T1I[bo] = ti;
}

// Inverse DFT y: thread per (b,o,x,y,kz).
__global__ void idfty(const float* __restrict__ T1R, const float* __restrict__ T1I,
                      float* __restrict__ T2R, float* __restrict__ T2I)
{
    const int id = blockIdx.x * blockDim.x + threadIdx.x;
    if (id >= BD * WIDTH_ * S1D * S2D * TMOD_) return;
    const int kz = id % TMOD_;
    const int y  = (id / TMOD_) % S2D;
    const int x  = (id / (S2D * TMOD_)) % S1D;
    const int o  = (id / (S1D * S2D * TMOD_)) % WIDTH_;
    const int b  = id / (WIDTH_ * S1D * S2D * TMOD_);
    const float* arp = T1R + (long)o * (BD * S1D * MX23 * TMOD_)
                     + ((long)(b * S1D + x) * MX23) * TMOD_ + kz;
    const float* aip = T1I + (long)o * (BD * S1D * MX23 * TMOD_)
                     + ((long)(b * S1D + x) * MX23) * TMOD_ + kz;
    float tr = 0.0f, ti = 0.0f;
#pragma unroll
    for (int my = 0; my < MX23; ++my) {
        float s, co;
        __sincosf((TWO_PI / S2D) * (float)((my - 11) * y), &s, &co);
        const float ar = arp[(long)my * TMOD_];
        const float ai = aip[(long)my * TMOD_];
        tr += ar * co - ai * s;
        ti += ai * co + ar * s;
    }
    const long bo = (long)o * (BD * S1D * S2D * TMOD_)
                  + ((long)(b * S1D + x) * S2D + y) * TMOD_ + kz;
    T2R[bo] = tr; T2I[bo] = ti;
}

// Inverse rfft z (hermitian half-spectrum, kz<8 nonzero). Thread per (b,o,x,y).
__global__ void idftz(const float* __restrict__ T2R, const float* __restrict__ T2I,
                      float* __restrict__ C)
{
    const int id = blockIdx.x * blockDim.x + threadIdx.x;
    if (id >= WIDTH_ * BD * S1D * S2D) return;
    const int o = id / (BD * S1D * S2D);
    const int r = id % (BD * S1D * S2D);
    const long bi = (long)o * (BD * S1D * S2D * TMOD_) + (long)r * TMOD_;
    float Rk[TMOD_], Ik[TMOD_];
#pragma unroll
    for (int k = 0; k < TMOD_; ++k) { Rk[k] = T2R[bi + k]; Ik[k] = T2I[bi + k]; }
    const float scale = 1.0f / (float)(S1D * S2D * S3D);
    float* cp = C + (long)o * NTOT + (long)r * S3D;
    for (int t = 0; t < S3D; ++t) {
        float v = Rk[0];
#pragma unroll
        for (int k = 1; k < TMOD_; ++k) {
            float s, co;
            __sincosf((TWO_PI / S3D) * (float)(k * t), &s, &co);
            v += 2.0f * (Rk[k] * co - Ik[k] * s);
        }
        cp[t] = v * scale;
    }
}

// ===========================================================================
// Q head second layer: out[n] = q_b2 + sum_o q_w2[o] * H[o][n]  (H pre-GELU'd)
// ===========================================================================
__global__ void q2_kernel(const float* __restrict__ H,
                          const float* __restrict__ w2, const float* __restrict__ b2,
                          float* __restrict__ out, int ncols)
{
    const int j = blockIdx.x * blockDim.x + threadIdx.x;
    if (j >= ncols) return;
    float acc = b2[0];
    const float* hp = H + j;
    for (int o = 0; o < 4 * WIDTH_; ++o) { acc += w2[o] * (*hp); hp += ncols; }
    out[j] = acc;
}

// ===========================================================================
extern "C" void kernel_launch(void* const* d_in, const int* in_sizes, int n_in,
                              void* d_out, int out_size, void* d_ws, size_t ws_size,
                              hipStream_t stream)
{
    const float* x_in = (const float*)d_in[0];
    const float* p_w  = (const float*)d_in[1];
    const float* p_b  = (const float*)d_in[2];
    const float* WLC  = (const float*)d_in[3];   // (4,64,64,12,1,8)
    const float* WLR  = (const float*)d_in[4];   // (4,64,64,11,11,8)
    const float* mw1  = (const float*)d_in[5];
    const float* mb1  = (const float*)d_in[6];
    const float* mw2  = (const float*)d_in[7];
    const float* mb2  = (const float*)d_in[8];
    const float* ww   = (const float*)d_in[9];
    const float* wb   = (const float*)d_in[10];
    const float* qw1  = (const float*)d_in[11];  // (256,64)
    const float* qb1  = (const float*)d_in[12];
    const float* qw2  = (const float*)d_in[13];  // (1,256)
    const float* qb2  = (const float*)d_in[14];

    const int N = NTOT;
    float* ws = (float*)d_ws;
    size_t off = 0;
    auto alloc = [&](size_t n) { float* p = ws + off; off += n; return p; };

    float* A   = alloc((size_t)WIDTH_ * N);            // current activation [c][n]
    float* B2  = alloc((size_t)WIDTH_ * N);            // skip-conv output
    float* C   = alloc((size_t)WIDTH_ * N);            // spectral conv real output
    float* D   = alloc((size_t)WIDTH_ * N);            // MLP hidden (reused as Q hidden)
    const size_t zsz = (size_t)WIDTH_ * BD * S1D * S2D * TMOD_;   // 4.19M
    float* ZR  = alloc(zsz);  float* ZI = alloc(zsz);  // z-DFT out / T2 reuse
    const size_t ysz = (size_t)WIDTH_ * BD * S1D * MX23 * TMOD_;  // 1.51M
    float* YR  = alloc(ysz);  float* YI = alloc(ysz);
    const size_t xsz = (size_t)WIDTH_ * BD * NMODE;               // 0.54M
    float* XSR = alloc(xsz);  float* XSI = alloc(xsz);
    float* SR  = alloc(xsz);  float* SI  = alloc(xsz);
    float* T1R = alloc(ysz);  float* T1I = alloc(ysz);

    const dim3 blk(256);
    lift_kernel<<<(N + 255) / 256, blk, 0, stream>>>(x_in, p_w, p_b, A);

    for (int l = 0; l < 4; ++l) {
        const long lcOff = (long)l * WIDTH_ * WIDTH_ * MODES_ * TMOD_;
        const long lrOff = (long)l * WIDTH_ * WIDTH_ * 11 * 11 * TMOD_;

        dftz_fwd<<<(WIDTH_ * BD * S1D * S2D) / 256, blk, 0, stream>>>(A, ZR, ZI);
        dfty_fwd<<<(WIDTH_ * BD * S1D * TMOD_) / 256, blk, 0, stream>>>(ZR, ZI, YR, YI);
        dftx_fwd<<<(WIDTH_ * BD * MX23 * TMOD_) / 256, blk, 0, stream>>>(YR, YI, XSR, XSI);
        mode_einsum<<<(BD * WIDTH_ * NMODE + 255) / 256, blk, 0, stream>>>(
            XSR, XSI, WLC + lcOff, WLR + lrOff, SR, SI);
        idftx<<<(BD * WIDTH_ * S1D * MX23 * TMOD_) / 256, blk, 0, stream>>>(SR, SI, T1R, T1I);
        idfty<<<(BD * WIDTH_ * S1D * S2D * TMOD_) / 256, blk, 0, stream>>>(T1R, T1I, ZR, ZI);
        idftz<<<(WIDTH_ * BD * S1D * S2D) / 256, blk, 0, stream>>>(ZR, ZI, C);

        // D = gelu(mlp_w1 * C + b1)
        gemm_wmma_k64<4><<<N / 128, blk, 0, stream>>>(
            mw1 + l * 4096, mb1 + l * 64, C, N, nullptr, 0, D, N, N, 2);
        // B2 = w_w * A + w_b
        gemm_wmma_k64<4><<<N / 128, blk, 0, stream>>>(
            ww + l * 4096, wb + l * 64, A, N, nullptr, 0, B2, N, N, 0);
        // A = [relu]( mlp_w2 * D + b2 + B2 )
        gemm_wmma_k64<4><<<N / 128, blk, 0, stream>>>(
            mw2 + l * 4096, mb2 + l * 64, D, N, B2, N, A, N, N, (l < 3) ? 1 : 0);
    }

    // Q head: 64 -> 256 (gelu, WMMA, N-chunked to reuse D) -> 1
    const int NC = 65536;   // 327680 = 5 * 65536
    for (int ch = 0; ch < N / NC; ++ch) {
        const long n_off = (long)ch * NC;
        gemm_wmma_k64<8><<<NC / 128, blk, 0, stream>>>(
            qw1, qb1, A + n_off, N, nullptr, 0, D, NC, NC, 2);
        gemm_wmma_k64<8><<<NC / 128, blk, 0, stream>>>(
            qw1 + 128 * 64, qb1 + 128, A + n_off, N, nullptr, 0,
            D + (long)128 * NC, NC, NC, 2);
        q2_kernel<<<NC / 256, blk, 0, stream>>>(D, qw2, qb2, (float*)d_out + n_off, NC);
    }
    (void)in_sizes; (void)n_in; (void)out_size; (void)ws_size;
}